// GCN_29222957482616
// MI455X (gfx1250) — compile-verified
//
#include <hip/hip_runtime.h>
#include <hip/hip_bf16.h>

typedef __attribute__((ext_vector_type(2))) float v2f;
typedef __attribute__((ext_vector_type(8))) float v8f;

#define N_NODES 50000
#define E_EDGES 800000
#define DIM 128
#define NF ((size_t)N_NODES * DIM)

// ---------------------------------------------------------------------------
// WMMA GEMM: C[M x 128] = op(A[M x 128]) @ B[128 x 128] (+ addvec broadcast)
// Block = 256 threads (8 waves). Each block: one 16-row tile of A staged in
// LDS; wave w computes the 16x16 tile at columns [16w, 16w+16).
// fp32 WMMA 16x16x4, 32 chained v_wmma per wave.
// ---------------------------------------------------------------------------
template <bool RELU_A, bool ADD_VEC>
__global__ void gcn_gemm_wmma(const float* __restrict__ A,
                              const float* __restrict__ B,
                              float* __restrict__ C,
                              const float* __restrict__ addvec) {
  __shared__ float lA[16 * DIM];  // 8 KB
  const int tidb = threadIdx.x;
  const int rowbase = blockIdx.x * 16;

  // Cooperative load of the 16x128 A tile (512 float4s across 256 threads).
  {
    const float4* srcp = (const float4*)(A + (size_t)rowbase * DIM);
    float4* dstp = (float4*)lA;
#pragma unroll
    for (int i = tidb; i < 512; i += 256) {
      float4 v = srcp[i];
      if (RELU_A) {
        v.x = fmaxf(v.x, 0.f); v.y = fmaxf(v.y, 0.f);
        v.z = fmaxf(v.z, 0.f); v.w = fmaxf(v.w, 0.f);
      }
      dstp[i] = v;
    }
  }
  __syncthreads();

  const int wave = tidb >> 5;          // 0..7 -> column tile
  const int lane = tidb & 31;
  const int mrow = lane & 15;          // M (for A) / N (for B,D) index
  const int khalf = (lane >> 4) * 2;   // K sub-offset: 0 or 2
  const int col = wave * 16 + mrow;

  v8f acc = {0.f, 0.f, 0.f, 0.f, 0.f, 0.f, 0.f, 0.f};
#pragma unroll
  for (int k0 = 0; k0 < DIM; k0 += 4) {
    const int ka = k0 + khalf;
    v2f av, bv;
    av.x = lA[mrow * DIM + ka];
    av.y = lA[mrow * DIM + ka + 1];
    bv.x = B[(size_t)ka * DIM + col];
    bv.y = B[(size_t)(ka + 1) * DIM + col];
    acc = __builtin_amdgcn_wmma_f32_16x16x4_f32(
        /*neg_a=*/false, av, /*neg_b=*/false, bv,
        /*c_mod=*/(short)0, acc, /*reuse_a=*/false, /*reuse_b=*/false);
  }

  const float addv = ADD_VEC ? addvec[col] : 0.f;
#pragma unroll
  for (int v = 0; v < 8; ++v) {
    const int r = rowbase + v + (lane >> 4) * 8;  // D: M = v + 8*(lane>>4)
    C[(size_t)r * DIM + col] = acc[v] + addv;
  }
}

// ---------------------------------------------------------------------------
// Degree / normalization helpers
// ---------------------------------------------------------------------------
__global__ void k_fill(float* __restrict__ p, float val, int n) {
  int t = blockIdx.x * blockDim.x + threadIdx.x;
  if (t < n) p[t] = val;
}

__global__ void k_deg_count(const int* __restrict__ dst, float* __restrict__ deg, int E) {
  int t = blockIdx.x * blockDim.x + threadIdx.x;
  if (t < E) atomicAdd(&deg[dst[t]], 1.0f);
}

__global__ void k_dinv(const float* __restrict__ deg, float* __restrict__ dinv, int n) {
  int t = blockIdx.x * blockDim.x + threadIdx.x;
  if (t < n) {
    float d = deg[t];
    dinv[t] = d > 0.f ? rsqrtf(d) : 0.f;
  }
}

// AGG[i][f] = H[i][f] * dinv[i]^2 (+ bias[f])   -- self-loop term + bias init
__global__ void k_init_agg(const float* __restrict__ H, const float* __restrict__ dinv,
                           const float* __restrict__ bias, float* __restrict__ AGG, int n) {
  int t = blockIdx.x * blockDim.x + threadIdx.x;
  if (t < n * DIM) {
    int i = t >> 7;
    int f = t & (DIM - 1);
    float di = dinv[i];
    float bv = bias ? bias[f] : 0.f;
    AGG[t] = H[t] * di * di + bv;
  }
}

// Scatter-add: 32 lanes per edge, float4 gather, 4 f32 atomics per lane.
__global__ void k_edge_agg(const float* __restrict__ H, float* __restrict__ AGG,
                           const int* __restrict__ src, const int* __restrict__ dst,
                           const float* __restrict__ dinv, int E) {
  int gtid = blockIdx.x * blockDim.x + threadIdx.x;
  int lane = gtid & 31;
  int e = gtid >> 5;
  int estride = (gridDim.x * blockDim.x) >> 5;
  for (; e < E; e += estride) {
    int s = src[e];
    int d = dst[e];
    float c = dinv[s] * dinv[d];
    float4 h = ((const float4*)(H + (size_t)s * DIM))[lane];
    float* ap = AGG + (size_t)d * DIM + lane * 4;
    atomicAdd(ap + 0, h.x * c);
    atomicAdd(ap + 1, h.y * c);
    atomicAdd(ap + 2, h.z * c);
    atomicAdd(ap + 3, h.w * c);
  }
}

// rvec[j] = sum_k relu(posts[root][k]) * W2[k][j]   (constant row of layer 2)
__global__ void k_rvec(const float* __restrict__ posts, const int* __restrict__ root,
                       const float* __restrict__ W2, float* __restrict__ rvec) {
  int j = threadIdx.x;  // 128 threads
  int r = *root;
  const float* pr = posts + (size_t)r * DIM;
  float s = 0.f;
#pragma unroll 4
  for (int k = 0; k < DIM; ++k) s += fmaxf(pr[k], 0.f) * W2[(size_t)k * DIM + j];
  rvec[j] = s;
}

// out[0:128] = AGG1[root]; out[128:256] = 0 (mean accumulator init)
__global__ void k_root_copy(const float* __restrict__ AGG1, const int* __restrict__ root,
                            float* __restrict__ out) {
  int f = threadIdx.x;  // 128 threads
  int r = *root;
  out[f] = AGG1[(size_t)r * DIM + f];
  out[DIM + f] = 0.f;
}

// out[128+f] += (1/N) * sum_i relu(AGG2[i][f] + b2[f])
__global__ void k_mean(const float* __restrict__ AGG2, const float* __restrict__ b2,
                       float* __restrict__ out, int N) {
  int f = threadIdx.x;  // 128 threads, 128 blocks
  float bf = b2[f];
  float s = 0.f;
  for (int i = blockIdx.x; i < N; i += gridDim.x)
    s += fmaxf(AGG2[(size_t)i * DIM + f] + bf, 0.f);
  atomicAdd(&out[DIM + f], s * (1.0f / N));
}

// ---------------------------------------------------------------------------
extern "C" void kernel_launch(void* const* d_in, const int* in_sizes, int n_in,
                              void* d_out, int out_size, void* d_ws, size_t ws_size,
                              hipStream_t stream) {
  (void)n_in; (void)out_size; (void)ws_size;
  const float* posts = (const float*)d_in[0];        // [N,128]
  const int* eidx = (const int*)d_in[1];             // [2,E]
  const int* root = (const int*)d_in[2];             // scalar
  const float* W1 = (const float*)d_in[3];           // [128,128]
  const float* b1 = (const float*)d_in[4];           // [128]
  const float* W2 = (const float*)d_in[5];           // [256,128]
  const float* b2 = (const float*)d_in[6];           // [128]
  float* out = (float*)d_out;                        // [256]

  const int N = in_sizes[0] / DIM;                   // 50000
  const int E = in_sizes[1] / 2;                     // 800000
  const int* srcp = eidx;
  const int* dstp = eidx + E;

  float* ws = (float*)d_ws;
  float* bufA = ws;              // H1, then H2          (N*128 floats)
  float* bufB = ws + NF;         // AGG1, then AGG2      (N*128 floats)
  float* dinv = ws + 2 * NF;     // [N]
  float* deg  = dinv + N_NODES;  // [N]
  float* rvec = deg + N_NODES;   // [128]

  const int TPB = 256;
  const int nElemBlk = (N * DIM + TPB - 1) / TPB;
  const int nNodeBlk = (N + TPB - 1) / TPB;
  const int nEdgeBlk = (E + TPB - 1) / TPB;
  const int gemmBlk = N / 16;                        // 3125 (N % 16 == 0)
  const int edgeAggBlk = 4096;                       // grid-stride over edges

  // Degrees (with self-loop) and dinv = rsqrt(deg)
  k_fill<<<nNodeBlk, TPB, 0, stream>>>(deg, 1.0f, N);
  k_deg_count<<<nEdgeBlk, TPB, 0, stream>>>(dstp, deg, E);
  k_dinv<<<nNodeBlk, TPB, 0, stream>>>(deg, dinv, N);

  // Layer 1: H1 = X @ W1 ; AGG1 = scatter(H1) + b1
  gcn_gemm_wmma<false, false><<<gemmBlk, TPB, 0, stream>>>(posts, W1, bufA, nullptr);
  k_init_agg<<<nElemBlk, TPB, 0, stream>>>(bufA, dinv, b1, bufB, N);
  k_edge_agg<<<edgeAggBlk, TPB, 0, stream>>>(bufA, bufB, srcp, dstp, dinv, E);

  // conv1_root -> out[0:128]; zero mean accumulator
  k_root_copy<<<1, DIM, 0, stream>>>(bufB, root, out);

  // Layer 2: constant part rvec = relu(posts[root]) @ W2[0:128];
  //          H2 = relu(AGG1) @ W2[128:256] + rvec ; AGG2 = scatter(H2)
  k_rvec<<<1, DIM, 0, stream>>>(posts, root, W2, rvec);
  gcn_gemm_wmma<true, true><<<gemmBlk, TPB, 0, stream>>>(bufB, W2 + DIM * DIM, bufA, rvec);
  k_init_agg<<<nElemBlk, TPB, 0, stream>>>(bufA, dinv, nullptr, bufB, N);
  k_edge_agg<<<edgeAggBlk, TPB, 0, stream>>>(bufA, bufB, srcp, dstp, dinv, E);

  // out[128:256] = mean_i relu(AGG2[i] + b2)
  k_mean<<<DIM, DIM, 0, stream>>>(bufB, b2, out, N);
}